// TTTM2BMMModule_13099650253057
// MI455X (gfx1250) — compile-verified
//
#include <hip/hip_runtime.h>

#define CS   16
#define F    64
#define HF   256
#define BNH  64

typedef __attribute__((ext_vector_type(2))) float v2f;
typedef __attribute__((ext_vector_type(8))) float v8f;

__device__ __forceinline__ v8f wmma4(v2f a, v2f b, v8f c) {
  // D = A(16x4, f32) * B(4x16, f32) + C(16x16, f32)
  return __builtin_amdgcn_wmma_f32_16x16x4_f32(false, a, false, b, (short)0, c,
                                               false, false);
}

// A fragment: row-major S[M][K], leading dim ld, tile row offset mofs.
__device__ __forceinline__ v2f ldA(const float* S, int ld, int mofs, int kb, int lane) {
  int m = mofs + (lane & 15);
  int k = kb + ((lane >> 4) << 1);
  v2f a; a.x = S[m * ld + k]; a.y = S[m * ld + k + 1]; return a;
}
// A fragment from transposed storage: logical A[m][k] = S[k*ld + m]
__device__ __forceinline__ v2f ldAT(const float* S, int ld, int mofs, int kb, int lane) {
  int m = mofs + (lane & 15);
  int k = kb + ((lane >> 4) << 1);
  v2f a; a.x = S[k * ld + m]; a.y = S[(k + 1) * ld + m]; return a;
}
// B fragment: row-major S[K][N], leading dim ld, tile col offset nofs.
__device__ __forceinline__ v2f ldB(const float* S, int ld, int nofs, int kb, int lane) {
  int n = nofs + (lane & 15);
  int k = kb + ((lane >> 4) << 1);
  v2f b; b.x = S[k * ld + n]; b.y = S[(k + 1) * ld + n]; return b;
}
// B fragment from transposed storage: logical B[k][n] = S[n*ld + k]
__device__ __forceinline__ v2f ldBT(const float* S, int ld, int nofs, int kb, int lane) {
  int n = nofs + (lane & 15);
  int k = kb + ((lane >> 4) << 1);
  v2f b; b.x = S[n * ld + k]; b.y = S[n * ld + k + 1]; return b;
}

__device__ __forceinline__ float wred(float v) {
  #pragma unroll
  for (int m = 16; m >= 1; m >>= 1) v += __shfl_xor(v, m, 32);
  return v;
}

__device__ __forceinline__ float gelu_f(float x) {
  return 0.5f * x * (1.0f + tanhf(0.79788456f * (x + 0.044715f * x * x * x)));
}
__device__ __forceinline__ float dgelu_f(float x) {
  float t = tanhf(0.79788456f * x * (1.0f + 0.044715f * x * x));
  return 0.5f * x * ((1.0f - t * t) * (0.79788456f + 0.1070322243f * x * x))
       + 0.5f * (1.0f + t);
}

__global__ __launch_bounds__(256, 1)
void ttt_m2_kernel(const float* __restrict__ XQ, const float* __restrict__ XK,
                   const float* __restrict__ XV, const float* __restrict__ coeff,
                   const float* __restrict__ coeff_last,
                   const float* __restrict__ W1g, const float* __restrict__ b1g,
                   const float* __restrict__ W2g, const float* __restrict__ b2g,
                   const float* __restrict__ lnw, const float* __restrict__ lnb,
                   float* __restrict__ out, int n_chunks) {
  const int bh   = blockIdx.x;    // 0..63
  const int tid  = threadIdx.x;   // 0..255 (8 wave32)
  const int lane = tid & 31;
  const int wid  = tid >> 5;
  const int h    = bh & 15;       // head index (NH = 16)

  // Recurrent state lives in LDS for the entire scan (CDNA5: 320KB/WGP).
  __shared__ float sW1[F * HF];   // 64KB
  __shared__ float sW2[HF * F];   // 64KB
  __shared__ float sb1[HF];
  __shared__ float sb2[F];
  __shared__ float sGam[F], sBet[F];
  __shared__ float sXK[CS * F], sXQ[CS * F], sXV[CS * F];
  __shared__ float sZ1[CS * HF];      // Z1; later reused for X2_bar
  __shared__ float sX2[CS * HF];
  __shared__ float sdZ1[CS * HF];
  __shared__ float sZ2[CS * F];       // Z2; later reused for Z2_bar
  __shared__ float sdZ2[CS * F];
  __shared__ float sB1[CS * HF];      // b1_bar
  __shared__ float sB2[CS * F];       // b2_bar
  __shared__ float sA1[CS * CS], sA2[CS * CS];
  __shared__ float sPart[8 * CS * CS];  // partial-K tiles / split reductions
  __shared__ float sC[CS];
  __shared__ float sCL;

  float* sX2b = sZ1;   // X2_bar overlays Z1 (Z1 dead after dZ1 phase)

  for (int i = tid; i < F * HF; i += 256) sW1[i] = W1g[bh * F * HF + i];
  for (int i = tid; i < HF * F; i += 256) sW2[i] = W2g[bh * HF * F + i];
  for (int i = tid; i < HF; i += 256) sb1[i] = b1g[bh * HF + i];
  if (tid < F) { sb2[tid] = b2g[bh * F + tid];
                 sGam[tid] = lnw[h * F + tid]; sBet[tid] = lnb[h * F + tid]; }

  // ---- software pipeline: preload chunk 0 into registers -----------------
  float rq[4], rk[4], rv[4], rc = 0.f, rcl = 0.f;
  {
    const long base = (long)bh * (CS * F);
    #pragma unroll
    for (int j = 0; j < 4; ++j) {
      int i = tid + j * 256;
      rq[j] = XQ[base + i]; rk[j] = XK[base + i]; rv[j] = XV[base + i];
    }
    if (tid < CS) rc = coeff[(long)bh * CS + tid];
    if (tid == 0) rcl = coeff_last[bh];
  }
  __syncthreads();

  for (int c = 0; c < n_chunks; ++c) {
    // publish current chunk to LDS
    #pragma unroll
    for (int j = 0; j < 4; ++j) {
      int i = tid + j * 256;
      sXQ[i] = rq[j]; sXK[i] = rk[j]; sXV[i] = rv[j];
    }
    if (tid < CS) sC[tid] = rc;
    if (tid == 0) sCL = rcl;
    __syncthreads();

    // issue next chunk's global loads now; they drain during compute below
    if (c + 1 < n_chunks) {
      const long nb = ((long)(c + 1) * BNH + bh) * (CS * F);
      #pragma unroll
      for (int j = 0; j < 4; ++j) {
        int i = tid + j * 256;
        rq[j] = XQ[nb + i]; rk[j] = XK[nb + i]; rv[j] = XV[nb + i];
      }
      if (tid < CS) rc = coeff[((long)(c + 1) * BNH + bh) * CS + tid];
      if (tid == 0) rcl = coeff_last[(long)(c + 1) * BNH + bh];
      if (c + 2 < n_chunks) {   // pull chunk c+2 toward L2 (global_prefetch_b8)
        const long pb = ((long)(c + 2) * BNH + bh) * (CS * F);
        __builtin_prefetch(&XQ[pb + tid * 4], 0, 1);
        __builtin_prefetch(&XK[pb + tid * 4], 0, 1);
        __builtin_prefetch(&XV[pb + tid * 4], 0, 1);
      }
    }

    // ---- Z1 = xk @ W1 + b1 ; X2 = gelu(Z1). 16 N-tiles, 2 per wave.
    #pragma unroll
    for (int t = 0; t < 2; ++t) {
      int nofs = (wid * 2 + t) * 16;
      v8f acc = {};
      #pragma unroll
      for (int kb = 0; kb < F; kb += 4)
        acc = wmma4(ldA(sXK, F, 0, kb, lane), ldB(sW1, HF, nofs, kb, lane), acc);
      #pragma unroll
      for (int r = 0; r < 8; ++r) {
        int m = r + 8 * (lane >> 4), n = nofs + (lane & 15);
        float z = acc[r] + sb1[n];
        sZ1[m * HF + n] = z;
        sX2[m * HF + n] = gelu_f(z);
      }
    }
    __syncthreads();

    // ---- Z2 = X2 @ W2 + b2. 4 tiles, K=256 split between wave pairs.
    {
      int nt = wid & 3, kh = wid >> 2, nofs = nt * 16;
      v8f acc = {};
      for (int kb = kh * 128; kb < kh * 128 + 128; kb += 4)
        acc = wmma4(ldA(sX2, HF, 0, kb, lane), ldB(sW2, F, nofs, kb, lane), acc);
      if (kh == 1) {
        #pragma unroll
        for (int r = 0; r < 8; ++r) {
          int m = r + 8 * (lane >> 4), n = lane & 15;
          sPart[nt * 256 + m * 16 + n] = acc[r];
        }
      }
      __syncthreads();
      if (kh == 0) {
        #pragma unroll
        for (int r = 0; r < 8; ++r) {
          int m = r + 8 * (lane >> 4), nl = lane & 15, n = nofs + nl;
          sZ2[m * F + n] = acc[r] + sPart[nt * 256 + m * 16 + nl] + sb2[n];
        }
      }
    }
    __syncthreads();

    // ---- dZ2 = ln_fused_l2_bwd(Z2, xv-xk). One row per wave (2 passes).
    #pragma unroll
    for (int rr = 0; rr < 2; ++rr) {
      int row = wid + rr * 8;
      int c0 = lane * 2, c1 = lane * 2 + 1;
      float x0 = sZ2[row * F + c0], x1 = sZ2[row * F + c1];
      float mu = wred(x0 + x1) * (1.0f / 64.0f);
      float d0 = x0 - mu, d1 = x1 - mu;
      float var = wred(d0 * d0 + d1 * d1) * (1.0f / 64.0f);
      float rstd = rsqrtf(var + 1e-6f);
      float xh0 = d0 * rstd, xh1 = d1 * rstd;
      float g0 = sGam[c0], g1 = sGam[c1];
      float tg0 = sXV[row * F + c0] - sXK[row * F + c0];
      float tg1 = sXV[row * F + c1] - sXK[row * F + c1];
      float go0 = (g0 * xh0 + sBet[c0]) - tg0;
      float go1 = (g1 * xh1 + sBet[c1]) - tg1;
      float gx0 = go0 * g0, gx1 = go1 * g1;
      float s1 = wred(gx0 + gx1);
      float s2 = wred(gx0 * xh0 + gx1 * xh1);
      float k = rstd * (1.0f / 64.0f);
      sdZ2[row * F + c0] = (64.0f * gx0 - s1 - xh0 * s2) * k;
      sdZ2[row * F + c1] = (64.0f * gx1 - s1 - xh1 * s2) * k;
    }
    __syncthreads();

    // ---- dZ1 = (dZ2 @ W2^T) * diff_gelu(Z1). 16 tiles, 2 per wave.
    #pragma unroll
    for (int t = 0; t < 2; ++t) {
      int nofs = (wid * 2 + t) * 16;
      v8f acc = {};
      #pragma unroll
      for (int kb = 0; kb < F; kb += 4)
        acc = wmma4(ldA(sdZ2, F, 0, kb, lane), ldBT(sW2, F, nofs, kb, lane), acc);
      #pragma unroll
      for (int r = 0; r < 8; ++r) {
        int m = r + 8 * (lane >> 4), n = nofs + (lane & 15);
        sdZ1[m * HF + n] = acc[r] * dgelu_f(sZ1[m * HF + n]);
      }
    }
    __syncthreads();

    // ---- b1_bar / b2_bar: per-column causal cumsum of dZ1 / dZ2.
    {
      float acc = 0.f;
      #pragma unroll
      for (int i = 0; i < CS; ++i) {
        acc += sdZ1[i * HF + tid];
        sB1[i * HF + tid] = sb1[tid] - sC[i] * acc;
      }
    }
    if (tid < F) {
      float acc = 0.f;
      #pragma unroll
      for (int i = 0; i < CS; ++i) {
        acc += sdZ2[i * F + tid];
        sB2[i * F + tid] = sb2[tid] - sC[i] * acc;
      }
    }
    // ---- A1 = tril(xq @ xk^T): 4 waves split K=64.
    if (wid < 4) {
      v8f acc = {};
      #pragma unroll
      for (int kb = wid * 16; kb < wid * 16 + 16; kb += 4)
        acc = wmma4(ldA(sXQ, F, 0, kb, lane), ldBT(sXK, F, 0, kb, lane), acc);
      #pragma unroll
      for (int r = 0; r < 8; ++r) {
        int m = r + 8 * (lane >> 4), n = lane & 15;
        sPart[wid * 256 + m * 16 + n] = acc[r];
      }
    }
    __syncthreads();
    {
      int m = tid >> 4, n = tid & 15;
      float s = sPart[tid] + sPart[256 + tid] + sPart[512 + tid] + sPart[768 + tid];
      sA1[tid] = (n <= m) ? s : 0.f;
    }
    __syncthreads();

    // ---- Z1_bar = xq@W1 - c*(A1@dZ1) + b1_bar ; X2_bar = gelu(Z1_bar).
    #pragma unroll
    for (int t = 0; t < 2; ++t) {
      int nofs = (wid * 2 + t) * 16;
      v8f a1 = {}, a2 = {};
      #pragma unroll
      for (int kb = 0; kb < F; kb += 4)
        a1 = wmma4(ldA(sXQ, F, 0, kb, lane), ldB(sW1, HF, nofs, kb, lane), a1);
      #pragma unroll
      for (int kb = 0; kb < CS; kb += 4)
        a2 = wmma4(ldA(sA1, CS, 0, kb, lane), ldB(sdZ1, HF, nofs, kb, lane), a2);
      #pragma unroll
      for (int r = 0; r < 8; ++r) {
        int m = r + 8 * (lane >> 4), n = nofs + (lane & 15);
        float z = a1[r] - sC[m] * a2[r] + sB1[m * HF + n];
        sX2b[m * HF + n] = gelu_f(z);
      }
    }
    __syncthreads();

    // ---- A2 = tril(X2_bar @ X2^T): 8 waves split K=256.
    {
      v8f acc = {};
      #pragma unroll
      for (int kb = wid * 32; kb < wid * 32 + 32; kb += 4)
        acc = wmma4(ldA(sX2b, HF, 0, kb, lane), ldBT(sX2, HF, 0, kb, lane), acc);
      #pragma unroll
      for (int r = 0; r < 8; ++r) {
        int m = r + 8 * (lane >> 4), n = lane & 15;
        sPart[wid * 256 + m * 16 + n] = acc[r];
      }
    }
    __syncthreads();
    {
      int m = tid >> 4, n = tid & 15;
      float s = 0.f;
      #pragma unroll
      for (int w = 0; w < 8; ++w) s += sPart[w * 256 + tid];
      sA2[tid] = (n <= m) ? s : 0.f;
    }
    __syncthreads();

    // ---- Z2_bar = X2_bar@W2 - c*(A2@dZ2) + b2_bar. 4 tiles, K split.
    {
      int nt = wid & 3, kh = wid >> 2, nofs = nt * 16;
      v8f acc = {};
      for (int kb = kh * 128; kb < kh * 128 + 128; kb += 4)
        acc = wmma4(ldA(sX2b, HF, 0, kb, lane), ldB(sW2, F, nofs, kb, lane), acc);
      if (kh == 1) {
        #pragma unroll
        for (int r = 0; r < 8; ++r) {
          int m = r + 8 * (lane >> 4), n = lane & 15;
          sPart[nt * 256 + m * 16 + n] = acc[r];
        }
      }
      v8f a2 = {};
      if (kh == 0) {
        #pragma unroll
        for (int kb = 0; kb < CS; kb += 4)
          a2 = wmma4(ldA(sA2, CS, 0, kb, lane), ldB(sdZ2, F, nofs, kb, lane), a2);
      }
      __syncthreads();
      if (kh == 0) {
        #pragma unroll
        for (int r = 0; r < 8; ++r) {
          int m = r + 8 * (lane >> 4), nl = lane & 15, n = nofs + nl;
          sZ2[m * F + n] = acc[r] + sPart[nt * 256 + m * 16 + nl]
                         - sC[m] * a2[r] + sB2[m * F + n];
        }
      }
    }
    __syncthreads();

    // ---- emit Z2_bar: out[bh, c*CS + row, col]
    for (int i = tid; i < CS * F; i += 256)
      out[(long)bh * n_chunks * CS * F + (long)c * CS * F + i] = sZ2[i];

    // ---- W1 -= cl * xk^T @ dZ1 (64 tiles: 4 mt x 16 nt; 8 per wave)
    #pragma unroll
    for (int i = 0; i < 8; ++i) {
      int t = wid * 8 + i, mt = t >> 4, nt = t & 15;
      v8f acc = {};
      #pragma unroll
      for (int kb = 0; kb < CS; kb += 4)
        acc = wmma4(ldAT(sXK, F, mt * 16, kb, lane),
                    ldB(sdZ1, HF, nt * 16, kb, lane), acc);
      #pragma unroll
      for (int r = 0; r < 8; ++r) {
        int m = mt * 16 + r + 8 * (lane >> 4), n = nt * 16 + (lane & 15);
        sW1[m * HF + n] -= sCL * acc[r];
      }
    }
    // ---- W2 -= cl * X2^T @ dZ2 (64 tiles: 16 mt x 4 nt; 8 per wave)
    #pragma unroll
    for (int i = 0; i < 8; ++i) {
      int t = wid * 8 + i, mt = t >> 2, nt = t & 3;
      v8f acc = {};
      #pragma unroll
      for (int kb = 0; kb < CS; kb += 4)
        acc = wmma4(ldAT(sX2, HF, mt * 16, kb, lane),
                    ldB(sdZ2, F, nt * 16, kb, lane), acc);
      #pragma unroll
      for (int r = 0; r < 8; ++r) {
        int m = mt * 16 + r + 8 * (lane >> 4), n = nt * 16 + (lane & 15);
        sW2[m * F + n] -= sCL * acc[r];
      }
    }
    // ---- carry biases: last row of the bar matrices
    if (tid < HF) sb1[tid] = sB1[15 * HF + tid];
    if (tid < F)  sb2[tid] = sB2[15 * F + tid];
    __syncthreads();
  }
}

extern "C" void kernel_launch(void* const* d_in, const int* in_sizes, int n_in,
                              void* d_out, int out_size, void* d_ws, size_t ws_size,
                              hipStream_t stream) {
  (void)n_in; (void)out_size; (void)d_ws; (void)ws_size;
  const float* XQ  = (const float*)d_in[0];
  const float* XK  = (const float*)d_in[1];
  const float* XV  = (const float*)d_in[2];
  const float* cf  = (const float*)d_in[3];
  const float* cfl = (const float*)d_in[4];
  const float* W1  = (const float*)d_in[5];
  const float* b1  = (const float*)d_in[6];
  const float* W2  = (const float*)d_in[7];
  const float* b2  = (const float*)d_in[8];
  const float* lnw = (const float*)d_in[9];
  const float* lnb = (const float*)d_in[10];
  int n_chunks = in_sizes[0] / (BNH * CS * F);   // = 128
  ttt_m2_kernel<<<BNH, 256, 0, stream>>>(XQ, XK, XV, cf, cfl, W1, b1, W2, b2,
                                         lnw, lnb, (float*)d_out, n_chunks);
}